// CompletionNet_42949672960764
// MI455X (gfx1250) — compile-verified
//
#include <hip/hip_runtime.h>

// CDNA5 / gfx1250 sparse-conv pipeline:
//   stage1: scalar scatter (outer product [M,1]x[1,16])   -> h_acc
//   bn+elu (in place)                                      -> h
//   stage2: V_WMMA_F32_16X16X4_F32 gather-GEMM-scatter     -> acc
//   bn+elu                                                 -> d_out

typedef __attribute__((ext_vector_type(2))) float v2f;
typedef __attribute__((ext_vector_type(8))) float v8f;

#define BN_EPS 1e-5f

__device__ __forceinline__ void atomic_add_f32(float* p, float v) {
    // native global_atomic_add_f32 (no CAS loop)
    unsafeAtomicAdd(p, v);
}

// ---------------- Stage 1: SparseConv(1 -> 16) scatter ----------------
__global__ void conv1_scatter(const float* __restrict__ x,
                              const float* __restrict__ W1,   // [K,1,16]
                              const int*   __restrict__ in_idx,
                              const int*   __restrict__ out_idx,
                              float*       __restrict__ h_acc, // [N,16]
                              int M, int KM) {
    int t = blockIdx.x * blockDim.x + threadIdx.x;
    if (t >= KM) return;
    int k = t / M;
    float xv = x[in_idx[t]];
    int ro = out_idx[t] * 16;
    const float* w = W1 + k * 16;
#pragma unroll
    for (int c = 0; c < 16; ++c)
        atomic_add_f32(&h_acc[ro + c], xv * w[c]);
}

// ---------------- Fused inference BatchNorm + ELU ----------------
__global__ void bn_elu(const float* __restrict__ in, float* __restrict__ out,
                       const float* __restrict__ g,  const float* __restrict__ b,
                       const float* __restrict__ mn, const float* __restrict__ vr,
                       int total, int Cmask) {
    int t = blockIdx.x * blockDim.x + threadIdx.x;
    if (t >= total) return;
    int c = t & Cmask;                       // C is a power of two (16 / 32)
    float s = g[c] * rsqrtf(vr[c] + BN_EPS);
    float y = (in[t] - mn[c]) * s + b[c];
    out[t] = (y > 0.0f) ? y : (expf(y) - 1.0f);  // ELU, alpha=1
}

// ---------------- Stage 2: SparseConv(16 -> 32) via f32 WMMA ----------------
// One wave handles (offset k, 16 gathered rows). Accumulates two 16x16 f32
// tiles (Cout 0..15 and 16..31) with 4 chained V_WMMA_F32_16X16X4_F32 each,
// then scatter-adds the D tiles with f32 atomics.
__global__ void conv2_wmma(const float* __restrict__ h,     // [N,16]
                           const float* __restrict__ W2,    // [K,16,32]
                           const int*   __restrict__ in_idx,
                           const int*   __restrict__ out_idx,
                           float*       __restrict__ acc,   // [N,32]
                           int M, int Mt, int totalWaves) {
    int wave = blockIdx.x * (blockDim.x >> 5) + (threadIdx.x >> 5);
    if (wave >= totalWaves) return;          // wave-uniform: EXEC stays all-ones
    int lane = threadIdx.x & 31;
    int half = lane >> 4;                    // 0: lanes 0-15, 1: lanes 16-31
    int l15  = lane & 15;

    int k  = wave / Mt;
    int m0 = (wave - k * Mt) << 4;
    int base = k * M + m0;                   // index into [K,M] maps (< 2^31)

    // A gather: lane covers row (m0 + l15); VGPR0/1 = channels (4q+2*half)+0/1
    int arow = in_idx[base + l15];
    const float* hp = h + arow * 16 + (half << 1);

    const float* wk = W2 + k * 512;          // 16*32 weights for this offset

    v8f c0 = {};                             // Cout 0..15
    v8f c1 = {};                             // Cout 16..31

#pragma unroll
    for (int q = 0; q < 4; ++q) {            // K = 16 = 4 chunks of 4
        v2f a;
        a.x = hp[4 * q + 0];
        a.y = hp[4 * q + 1];
        int ci0 = 4 * q + (half << 1);       // B: lanes16-31 carry K rows 2,3
        v2f b0, b1;
        b0.x = wk[(ci0    ) * 32 +      l15];
        b0.y = wk[(ci0 + 1) * 32 +      l15];
        b1.x = wk[(ci0    ) * 32 + 16 + l15];
        b1.y = wk[(ci0 + 1) * 32 + 16 + l15];
        c0 = __builtin_amdgcn_wmma_f32_16x16x4_f32(false, a, false, b0,
                                                   (short)0, c0, false, false);
        c1 = __builtin_amdgcn_wmma_f32_16x16x4_f32(false, a, false, b1,
                                                   (short)0, c1, false, false);
    }

    // D layout: lane<16 -> M=v ; lane>=16 -> M=v+8 ; N = l15 (+16 for c1)
    int rows[8];
#pragma unroll
    for (int v = 0; v < 8; ++v)
        rows[v] = out_idx[base + (half << 3) + v] * 32;
#pragma unroll
    for (int v = 0; v < 8; ++v) {
        atomic_add_f32(&acc[rows[v] +      l15], c0[v]);
        atomic_add_f32(&acc[rows[v] + 16 + l15], c1[v]);
    }
}

// Scalar tail for M not divisible by 16 (not hit for M=200000, kept for safety)
__global__ void conv2_scalar_tail(const float* __restrict__ h,
                                  const float* __restrict__ W2,
                                  const int*   __restrict__ in_idx,
                                  const int*   __restrict__ out_idx,
                                  float*       __restrict__ acc,
                                  int M, int mStart, int K) {
    int rem = M - mStart;
    int total = K * rem * 32;
    int t = blockIdx.x * blockDim.x + threadIdx.x;
    if (t >= total) return;
    int co = t & 31;
    int km = t >> 5;
    int k  = km / rem;
    int m  = mStart + (km - k * rem);
    int e  = k * M + m;
    const float* hr = h + in_idx[e] * 16;
    const float* wp = W2 + k * 512 + co;
    float s = 0.0f;
#pragma unroll
    for (int ci = 0; ci < 16; ++ci) s += hr[ci] * wp[ci * 32];
    atomic_add_f32(&acc[out_idx[e] * 32 + co], s);
}

extern "C" void kernel_launch(void* const* d_in, const int* in_sizes, int n_in,
                              void* d_out, int out_size, void* d_ws, size_t ws_size,
                              hipStream_t stream) {
    const float* x      = (const float*)d_in[0];
    const float* W1     = (const float*)d_in[1];
    const float* g1     = (const float*)d_in[2];
    const float* b1     = (const float*)d_in[3];
    const float* m1     = (const float*)d_in[4];
    const float* v1     = (const float*)d_in[5];
    const float* W2     = (const float*)d_in[6];
    const float* g2     = (const float*)d_in[7];
    const float* b2     = (const float*)d_in[8];
    const float* m2     = (const float*)d_in[9];
    const float* v2     = (const float*)d_in[10];
    const int*   in_idx  = (const int*)d_in[11];
    const int*   out_idx = (const int*)d_in[12];

    const int N  = in_sizes[0];            // 400000
    const int K  = in_sizes[1] / 16;       // 27 (W1 is [K,1,16])
    const int KM = in_sizes[11];           // K*M
    const int M  = KM / K;                 // 200000

    float* h_buf = (float*)d_ws;                       // [N,16]
    float* acc   = h_buf + (size_t)N * 16;             // [N,32]

    // zero both scatter accumulators (graph-capture safe)
    hipMemsetAsync(d_ws, 0, (size_t)N * 48 * sizeof(float), stream);

    // stage 1: SparseConv(1->16)
    conv1_scatter<<<(KM + 255) / 256, 256, 0, stream>>>(
        x, W1, in_idx, out_idx, h_buf, M, KM);

    // BN + ELU (in place on h)
    bn_elu<<<(N * 16 + 255) / 256, 256, 0, stream>>>(
        h_buf, h_buf, g1, b1, m1, v1, N * 16, 15);

    // stage 2: SparseConv(16->32) with f32 WMMA
    const int Mt = M / 16;
    const int totalWaves = K * Mt;
    if (totalWaves > 0) {
        const int blocks = (totalWaves + 7) / 8;   // 8 waves / 256-thread block
        conv2_wmma<<<blocks, 256, 0, stream>>>(
            h_buf, W2, in_idx, out_idx, acc, M, Mt, totalWaves);
    }
    const int mStart = Mt * 16;
    if (mStart < M) {
        const int total = K * (M - mStart) * 32;
        conv2_scalar_tail<<<(total + 255) / 256, 256, 0, stream>>>(
            h_buf, W2, in_idx, out_idx, acc, M, mStart, K);
    }

    // BN + ELU -> final output
    bn_elu<<<(N * 32 + 255) / 256, 256, 0, stream>>>(
        acc, (float*)d_out, g2, b2, m2, v2, N * 32, 31);
}